// FeatureMamba_2525440770258
// MI455X (gfx1250) — compile-verified
//
#include <hip/hip_runtime.h>

// ---------------------------------------------------------------------------
// FeatureMamba forward for MI455X (gfx1250, wave32).
// GEMMs: LDS-tiled, software-pipelined V_WMMA_F32_16X16X4_F32 (exact fp32).
// Block tile 128x64xBK16, 8 waves, each wave = 64x16 strip (4 accumulators).
// ---------------------------------------------------------------------------

namespace {

constexpr int kB      = 32;
constexpr int kSeq    = 1024;
constexpr int kDM     = 512;
constexpr int kNL     = 4;
constexpr int kDState = 16;
constexpr int kDConv  = 4;
constexpr int kDInner = 1024;
constexpr int kRows   = kB * kSeq;            // 32768 flattened (b, l) rows
constexpr float kEps  = 1.1920929e-07f;

typedef __attribute__((ext_vector_type(2))) float v2f;
typedef __attribute__((ext_vector_type(8))) float v8f;

// ---------------------------- device helpers ------------------------------

__device__ __forceinline__ float silu_f(float x) {
  return x / (1.0f + expf(-x));
}

__device__ __forceinline__ float gelu_f(float x) {
  // exact (erf-based) gelu, matches jax.nn.gelu(approximate=False)
  return 0.5f * x * (1.0f + erff(x * 0.70710678118654752440f));
}

__device__ __forceinline__ float softplus_f(float x) {
  // numerically stable log(1+exp(x))
  return fmaxf(x, 0.0f) + log1pf(expf(-fabsf(x)));
}

// efficient-kan cubic B-spline basis on uniform grid [-1,1], G=5, k=3 -> 8 bases
__device__ __forceinline__ void bspline8(float x, float* out) {
  const float h = 0.4f;
  float g[12];
#pragma unroll
  for (int j = 0; j < 12; j++) g[j] = (float)(j - 3) * h - 1.0f;
  float b[11];
#pragma unroll
  for (int j = 0; j < 11; j++) b[j] = (x >= g[j] && x < g[j + 1]) ? 1.0f : 0.0f;
#pragma unroll
  for (int k = 1; k <= 3; k++) {
    float inv = 1.0f / ((float)k * h);
#pragma unroll
    for (int j = 0; j < 11 - k; j++) {
      b[j] = (x - g[j]) * inv * b[j] + (g[j + k + 1] - x) * inv * b[j + 1];
    }
  }
#pragma unroll
  for (int j = 0; j < 8; j++) out[j] = b[j];
}

// ------------------------------- kernels ----------------------------------

// KAN input layer: in_features=1 -> DM. One wave per (b,l) row.
__global__ __launch_bounds__(256) void kan_in_kernel(
    const float* __restrict__ X,      // (kRows,) = x viewed (B,SEQ,1)
    const float* __restrict__ bw,     // (DM,1)
    const float* __restrict__ sw,     // (DM,1,8)
    const float* __restrict__ sc,     // (DM,1)
    float* __restrict__ H, float* __restrict__ RES2) {
  int wave = threadIdx.x >> 5, lane = threadIdx.x & 31;
  int row = blockIdx.x * 8 + wave;
  if (row >= kRows) return;
  float x = X[row];
  float bs[8];
  bspline8(x, bs);
  float sx = silu_f(x);
  for (int d = lane; d < kDM; d += 32) {
    float acc = sx * bw[d];
    float s = sc[d];
#pragma unroll
    for (int k = 0; k < 8; k++) acc += bs[k] * sw[d * 8 + k] * s;
    size_t o = (size_t)row * kDM + d;
    H[o] = acc;
    RES2[o] = acc;
  }
}

// LDS-tiled fp32 WMMA GEMM:  C(M,N) = act(A(M,K) @ W(N,K)^T + bias)
// act: 0=none, 1=softplus, 2=gelu.
// Grid: (N/64, M/128). Block: 256 threads = 8 waves; wave (wm,wn) in 2x4,
// each wave computes a 64x16 strip as four 16x16 WMMA tiles.
// Requires M%128==0, N%64==0, K%16==0 (true for every call site).
__global__ __launch_bounds__(256) void gemm_f32_wmma(
    const float* __restrict__ A, int lda,
    const float* __restrict__ W,                 // (N,K), row-major
    const float* __restrict__ bias,              // (N,) or nullptr
    float* __restrict__ C, int ldc,
    int K, int act) {
  constexpr int BM = 128, BN = 64, BK = 16, LDP = BK + 4;  // pad -> 16B aligned
  __shared__ float As[BM * LDP];
  __shared__ float Ws[BN * LDP];

  const int tid  = threadIdx.x;
  const int lane = tid & 31;
  const int wave = tid >> 5;
  const int wm   = wave >> 2;        // 0..1 : 64-row half
  const int wn   = wave & 3;         // 0..3 : 16-col strip
  const int mrow = lane & 15;
  const int half = lane >> 4;        // frag K-pair select: 0 -> K{0,1}, 1 -> K{2,3}
  const int kh   = half << 1;

  const size_t rbase = (size_t)blockIdx.y * BM;
  const size_t nbase = (size_t)blockIdx.x * BN;

  // staging slots: A tile 128x16 = 512 float4 (2/thread), W tile 64x16 = 256 (1/thread)
  const int srow = tid >> 2;
  const int sc4  = (tid & 3) << 2;

  // prologue: prefetch chunk 0 into registers
  float4 ra0 = *(const float4*)(A + (rbase + srow) * (size_t)lda + sc4);
  float4 ra1 = *(const float4*)(A + (rbase + srow + 64) * (size_t)lda + sc4);
  float4 rw  = *(const float4*)(W + (nbase + srow) * (size_t)K + sc4);

  v8f acc[4] = {v8f{}, v8f{}, v8f{}, v8f{}};
  const int nchunks = K / BK;
  for (int kc = 0; kc < nchunks; kc++) {
    __syncthreads();                               // previous compute done
    *(float4*)&As[srow * LDP + sc4]        = ra0;  // publish chunk kc
    *(float4*)&As[(srow + 64) * LDP + sc4] = ra1;
    *(float4*)&Ws[srow * LDP + sc4]        = rw;
    __syncthreads();
    if (kc + 1 < nchunks) {                        // prefetch chunk kc+1
      int kb = (kc + 1) * BK;
      ra0 = *(const float4*)(A + (rbase + srow) * (size_t)lda + kb + sc4);
      ra1 = *(const float4*)(A + (rbase + srow + 64) * (size_t)lda + kb + sc4);
      rw  = *(const float4*)(W + (nbase + srow) * (size_t)K + kb + sc4);
    }
#pragma unroll
    for (int kk = 0; kk < 4; kk++) {               // 4 wmma K-steps per chunk
      int kcol = (kk << 2) + kh;
      v2f bf = *(const v2f*)&Ws[(wn * 16 + mrow) * LDP + kcol];
#pragma unroll
      for (int s = 0; s < 4; s++) {
        v2f af = *(const v2f*)&As[(wm * 64 + s * 16 + mrow) * LDP + kcol];
        acc[s] = __builtin_amdgcn_wmma_f32_16x16x4_f32(
            false, af, false, bf, (short)0, acc[s], false, false);
      }
    }
  }

  const int col = (int)nbase + wn * 16 + mrow;
  const float bv = bias ? bias[col] : 0.0f;
#pragma unroll
  for (int s = 0; s < 4; s++) {
#pragma unroll
    for (int v = 0; v < 8; v++) {
      size_t row = rbase + (size_t)(wm * 64 + s * 16 + v + (half << 3));
      float val = acc[s][v] + bv;
      if (act == 1) val = softplus_f(val);
      else if (act == 2) val = gelu_f(val);
      C[row * (size_t)ldc + col] = val;
    }
  }
}

// Causal depthwise conv1d (d_conv=4) + bias + SiLU. xz cols [0, d_inner).
__global__ __launch_bounds__(256) void conv_silu_kernel(
    const float* __restrict__ XZ,     // (kRows, 2*d_inner)
    const float* __restrict__ cw,     // (d_inner, 1, 4)
    const float* __restrict__ cb,     // (d_inner,)
    float* __restrict__ XC) {         // (kRows, d_inner)
  size_t t = (size_t)blockIdx.x * blockDim.x + threadIdx.x;
  if (t >= (size_t)kRows * kDInner) return;
  int d = (int)(t % kDInner);
  size_t n = t / kDInner;
  int l = (int)(n % kSeq);
  float acc = cb[d];
#pragma unroll
  for (int i = 0; i < kDConv; i++) {
    int ls = l - (kDConv - 1) + i;
    if (ls >= 0) acc += XZ[(n - (size_t)(kDConv - 1) + i) * (2 * kDInner) + d] *
                        cw[d * kDConv + i];
  }
  XC[(size_t)n * kDInner + d] = silu_f(acc);
}

// Selective scan. block = 256 channels of one batch; loop over L with
// per-step B/C staged in LDS. Fuses +xc*D and *silu(z).
__global__ __launch_bounds__(256) void scan_kernel(
    const float* __restrict__ DT,     // (kRows, d_inner)  softplus'd dt
    const float* __restrict__ DBL,    // (kRows, 64)  [dt_raw | B | C]
    const float* __restrict__ XC,     // (kRows, d_inner)
    const float* __restrict__ XZ,     // (kRows, 2*d_inner) (z in cols 1024..2047)
    const float* __restrict__ Alog,   // (d_inner, 16)
    const float* __restrict__ Dp,     // (d_inner,)
    float* __restrict__ Y) {          // (kRows, d_inner)
  int b = blockIdx.y;
  int d = blockIdx.x * 256 + threadIdx.x;
  __shared__ float Bs[kDState], Cs[kDState];
  float A[kDState], h[kDState];
#pragma unroll
  for (int n = 0; n < kDState; n++) {
    A[n] = -expf(Alog[d * kDState + n]);
    h[n] = 0.0f;
  }
  float Dd = Dp[d];
  for (int l = 0; l < kSeq; l++) {
    size_t row = (size_t)b * kSeq + l;
    if (threadIdx.x < 16)
      Bs[threadIdx.x] = DBL[row * 64 + 32 + threadIdx.x];
    else if (threadIdx.x < 32)
      Cs[threadIdx.x - 16] = DBL[row * 64 + 48 + (threadIdx.x - 16)];
    __syncthreads();
    float dt = DT[row * kDInner + d];
    float xv = XC[row * kDInner + d];
    float zv = XZ[row * (2 * kDInner) + kDInner + d];
    float y = 0.0f;
#pragma unroll
    for (int n = 0; n < kDState; n++) {
      h[n] = h[n] * expf(dt * A[n]) + dt * xv * Bs[n];
      y += h[n] * Cs[n];
    }
    Y[row * kDInner + d] = (y + xv * Dd) * silu_f(zv);
    __syncthreads();
  }
}

// out[row] = rmsnorm(Aa[row] + Bb[row]) * w   (out may alias Aa or Bb)
template <int DIM>
__global__ __launch_bounds__(256) void rmsnorm_add_kernel(
    const float* __restrict__ Aa, const float* __restrict__ Bb,
    const float* __restrict__ w, float* __restrict__ out) {
  constexpr int NPT = DIM / 256;
  __shared__ float red[256];
  size_t row = blockIdx.x;
  float v[NPT];
  float ss = 0.0f;
#pragma unroll
  for (int i = 0; i < NPT; i++) {
    int idx = threadIdx.x + i * 256;
    float t = Aa[row * DIM + idx] + Bb[row * DIM + idx];
    v[i] = t;
    ss += t * t;
  }
  red[threadIdx.x] = ss;
  __syncthreads();
  for (int s = 128; s > 0; s >>= 1) {
    if (threadIdx.x < s) red[threadIdx.x] += red[threadIdx.x + s];
    __syncthreads();
  }
  float rms = 1.0f / sqrtf(red[0] / (float)DIM + kEps);
#pragma unroll
  for (int i = 0; i < NPT; i++) {
    int idx = threadIdx.x + i * 256;
    out[row * DIM + idx] = v[i] * rms * w[idx];
  }
}

// KAN output layer: DM -> 1. One block per row; block reduction.
__global__ __launch_bounds__(256) void kan_out_kernel(
    const float* __restrict__ E,      // (kRows, DM)
    const float* __restrict__ bw,     // (1, DM)
    const float* __restrict__ sw,     // (1, DM, 8)
    const float* __restrict__ sc,     // (1, DM)
    float* __restrict__ DEC) {        // (kRows,)
  __shared__ float red[256];
  size_t row = blockIdx.x;
  float acc = 0.0f;
  for (int d = threadIdx.x; d < kDM; d += 256) {
    float e = E[row * kDM + d];
    float bs[8];
    bspline8(e, bs);
    float t = silu_f(e) * bw[d];
    float s = sc[d];
#pragma unroll
    for (int k = 0; k < 8; k++) t += bs[k] * sw[d * 8 + k] * s;
    acc += t;
  }
  red[threadIdx.x] = acc;
  __syncthreads();
  for (int s = 128; s > 0; s >>= 1) {
    if (threadIdx.x < s) red[threadIdx.x] += red[threadIdx.x + s];
    __syncthreads();
  }
  if (threadIdx.x == 0) DEC[row] = red[0];
}

// ------------------------------ host side ---------------------------------

void gemm(const float* A, int lda, const float* W, const float* bias,
          float* C, int ldc, int M, int N, int K, int act, hipStream_t s) {
  dim3 grid(N / 64, M / 128);
  gemm_f32_wmma<<<grid, 256, 0, s>>>(A, lda, W, bias, C, ldc, K, act);
}

}  // namespace

extern "C" void kernel_launch(void* const* d_in, const int* in_sizes, int n_in,
                              void* d_out, int out_size, void* d_ws, size_t ws_size,
                              hipStream_t stream) {
  const float* x             = (const float*)d_in[0];
  const float* kan_in_bw     = (const float*)d_in[1];
  const float* kan_in_sw     = (const float*)d_in[2];
  const float* kan_in_sc     = (const float*)d_in[3];
  const float* kan_out_bw    = (const float*)d_in[4];
  const float* kan_out_sw    = (const float*)d_in[5];
  const float* kan_out_sc    = (const float*)d_in[6];
  const float* in_proj_w     = (const float*)d_in[7];
  const float* conv_w        = (const float*)d_in[8];
  const float* conv_b        = (const float*)d_in[9];
  const float* x_proj_w      = (const float*)d_in[10];
  const float* dt_proj_w     = (const float*)d_in[11];
  const float* dt_proj_b     = (const float*)d_in[12];
  const float* A_log         = (const float*)d_in[13];
  const float* D_param       = (const float*)d_in[14];
  const float* out_proj_w    = (const float*)d_in[15];
  const float* mlp_w1        = (const float*)d_in[16];
  const float* mlp_b1        = (const float*)d_in[17];
  const float* mlp_w2        = (const float*)d_in[18];
  const float* mlp_b2        = (const float*)d_in[19];
  const float* ln_w          = (const float*)d_in[20];
  const float* ln2_w         = (const float*)d_in[21];
  const float* lnx_w         = (const float*)d_in[22];
  float* out = (float*)d_out;

  // workspace layout (floats), kRows rows each
  float* ws   = (float*)d_ws;
  size_t R    = kRows;
  float* H    = ws;                  // (R, 512)  current hidden
  float* T    = H    + R * 512;      // (R, 512)  temp (mamba/mlp out, enc)
  float* RES2 = T    + R * 512;      // (R, 512)  outer residual
  float* XZ   = RES2 + R * 512;      // (R, 2048) in_proj out / mlp hidden
  float* XC   = XZ   + R * 2048;     // (R, 1024) conv+silu out
  float* DBL  = XC   + R * 1024;     // (R, 64)   x_proj out; reused as DEC
  float* DT   = DBL  + R * 64;       // (R, 1024) softplus(dt_proj)
  float* Y    = DT   + R * 1024;     // (R, 1024) scan output

  // 1) KAN input layer: x (B,SEQ,1) -> H (R, DM); also residual2
  kan_in_kernel<<<kRows / 8, 256, 0, stream>>>(x, kan_in_bw, kan_in_sw,
                                               kan_in_sc, H, RES2);

  for (int i = 0; i < kNL; i++) {
    const float* ipw = in_proj_w  + (size_t)i * 2 * kDInner * kDM;
    const float* cw  = conv_w     + (size_t)i * kDInner * kDConv;
    const float* cb  = conv_b     + (size_t)i * kDInner;
    const float* xpw = x_proj_w   + (size_t)i * 64 * kDInner;
    const float* dpw = dt_proj_w  + (size_t)i * kDInner * 32;
    const float* dpb = dt_proj_b  + (size_t)i * kDInner;
    const float* al  = A_log      + (size_t)i * kDInner * kDState;
    const float* dp  = D_param    + (size_t)i * kDInner;
    const float* opw = out_proj_w + (size_t)i * kDM * kDInner;

    // in_proj: (R,512) @ (2048,512)^T -> XZ (R,2048)
    gemm(H, kDM, ipw, nullptr, XZ, 2 * kDInner, kRows, 2 * kDInner, kDM, 0, stream);
    // depthwise causal conv + silu -> XC
    conv_silu_kernel<<<(int)(((size_t)kRows * kDInner + 255) / 256), 256, 0,
                       stream>>>(XZ, cw, cb, XC);
    // x_proj: (R,1024) @ (64,1024)^T -> DBL (R,64)
    gemm(XC, kDInner, xpw, nullptr, DBL, 64, kRows, 64, kDInner, 0, stream);
    // dt_proj + bias + softplus: (R,32) @ (1024,32)^T -> DT (R,1024)
    gemm(DBL, 64, dpw, dpb, DT, kDInner, kRows, kDInner, 32, 1, stream);
    // selective scan (fused D-skip + z-gate) -> Y
    {
      dim3 g(kDInner / 256, kB);
      scan_kernel<<<g, 256, 0, stream>>>(DT, DBL, XC, XZ, al, dp, Y);
    }
    // out_proj: (R,1024) @ (512,1024)^T -> T (R,512)
    gemm(Y, kDInner, opw, nullptr, T, kDM, kRows, kDM, kDInner, 0, stream);
    // mlp1 (gelu): (R,512) @ (1024,512)^T -> XZ (ld 1024, reuse)
    gemm(T, kDM, mlp_w1, mlp_b1, XZ, 2 * kDM, kRows, 2 * kDM, kDM, 2, stream);
    // mlp2 (gelu): (R,1024) @ (512,1024)^T -> T
    gemm(XZ, 2 * kDM, mlp_w2, mlp_b2, T, kDM, kRows, kDM, 2 * kDM, 2, stream);
    // h = rmsnorm(T + H) * ln_w   (in-place into H)
    rmsnorm_add_kernel<512><<<kRows, 256, 0, stream>>>(T, H, ln_w, H);
  }

  // enc = rmsnorm(H + RES2) * ln2_w -> T
  rmsnorm_add_kernel<512><<<kRows, 256, 0, stream>>>(H, RES2, ln2_w, T);
  // KAN output layer: T (R,512) -> DEC (R,)   (reuse DBL as DEC)
  float* DEC = DBL;
  kan_out_kernel<<<kRows, 256, 0, stream>>>(T, kan_out_bw, kan_out_sw,
                                            kan_out_sc, DEC);
  // final: out = rmsnorm(dec + x) * lnx_w over SEQ, per batch row
  rmsnorm_add_kernel<1024><<<kB, 256, 0, stream>>>(DEC, x, lnx_w, out);
}